// SelectiveSSMBlock_28449863368951
// MI455X (gfx1250) — compile-verified
//
#include <hip/hip_runtime.h>
#include <cstdint>
#include <cstddef>

// ---------------- problem constants ----------------
#define BATCH 8
#define PC    64
#define IMG   32
#define NF    256   // D: number of tokens (scan "channel" dim)
#define SL    256   // L: scan length
#define NST   256   // N: state dim
#define WAVES_PER_BLOCK 8
#define STEPS 16                 // steps staged per LDS block
#define NBLK  (SL / STEPS)       // 16

typedef float fvec4 __attribute__((ext_vector_type(4)));
typedef float v2f   __attribute__((ext_vector_type(2)));
typedef float v8f   __attribute__((ext_vector_type(8)));
typedef int   v4i   __attribute__((ext_vector_type(4)));

#if __has_builtin(__builtin_amdgcn_global_load_async_to_lds_b128)
#define HAVE_ASYNC_LDS 1
#endif
#if __has_builtin(__builtin_amdgcn_wmma_f32_16x16x4_f32)
#define HAVE_WMMA_F32X4 1
#endif

__device__ __forceinline__ float fast_exp2(float x) {
#if __has_builtin(__builtin_amdgcn_exp2f)
  return __builtin_amdgcn_exp2f(x);     // v_exp_f32
#else
  return exp2f(x);
#endif
}
__device__ __forceinline__ float fast_rcp(float x) {
#if __has_builtin(__builtin_amdgcn_rcpf)
  return __builtin_amdgcn_rcpf(x);      // v_rcp_f32 (no div macro)
#else
  return 1.f / x;
#endif
}
__device__ __forceinline__ float softplus_f(float v) {
  return (v > 20.f) ? v : __logf(1.f + __expf(v));
}
__device__ __forceinline__ float silu_f(float v) {
  return v * fast_rcp(1.f + __expf(-v));
}

// 16-byte global -> LDS copy; async (ASYNCcnt-tracked) on gfx1250.
__device__ __forceinline__ void copy16_to_lds(const float* g, float* l) {
#ifdef HAVE_ASYNC_LDS
  typedef __attribute__((address_space(1))) v4i* g4_t;
  typedef __attribute__((address_space(3))) v4i* l4_t;
  __builtin_amdgcn_global_load_async_to_lds_b128((g4_t)g, (l4_t)l, 0, 0);
#else
  *(fvec4*)l = *(const fvec4*)g;
#endif
}
__device__ __forceinline__ void wait_async_copies() {
#ifdef HAVE_ASYNC_LDS
#if __has_builtin(__builtin_amdgcn_s_wait_asynccnt)
  __builtin_amdgcn_s_wait_asynccnt(0);
#else
  asm volatile("s_wait_asynccnt 0x0" ::: "memory");
#endif
#endif
}
__device__ __forceinline__ void wave_fence() {
#if __has_builtin(__builtin_amdgcn_wave_barrier)
  __builtin_amdgcn_wave_barrier();
#endif
}

// ---------------------------------------------------------------------------
// Phase 1: transpose B_param / C_param from [b][n][l] to [b][l][n] so the per-
// step state columns become contiguous for async LDS staging.
// ---------------------------------------------------------------------------
__global__ __launch_bounds__(256)
void transpose_bc_kernel(const float* __restrict__ Bp, const float* __restrict__ Cp,
                         float* __restrict__ Bt, float* __restrict__ Ct) {
  __shared__ float tile[32][33];
  const int zb  = blockIdx.z;       // 0..15 : b*2 + which
  const int b   = zb >> 1;
  const bool isC = (zb & 1) != 0;
  const float* src = isC ? Cp : Bp;
  float*       dst = isC ? Ct : Bt;
  const int l0 = blockIdx.x * 32;
  const int n0 = blockIdx.y * 32;
  const int tx = threadIdx.x;       // 0..31
  const int ty = threadIdx.y;       // 0..7
  const size_t base = (size_t)b * NF * SL;
#pragma unroll
  for (int r = 0; r < 4; ++r)
    tile[ty + 8 * r][tx] = src[base + (size_t)(n0 + ty + 8 * r) * SL + (l0 + tx)];
  __syncthreads();
#pragma unroll
  for (int r = 0; r < 4; ++r)
    dst[base + (size_t)(l0 + ty + 8 * r) * NST + (n0 + tx)] = tile[tx][ty + 8 * r];
}

// ---------------------------------------------------------------------------
// Phase 2: selective scan. One wave per (b, d); 8 states per lane (n = 8*lane+k).
// 8 waves per workgroup share one batch index b, so B/C step-columns staged in
// LDS are reused 8x. y-reduction across lanes done 16 steps at a time with
// V_WMMA_F32_16X16X4_F32 (A = ones, B = per-lane partials).
// ---------------------------------------------------------------------------
__global__ __launch_bounds__(256)
void ssm_scan_kernel(const float* __restrict__ x,     const float* __restrict__ A,
                     const float* __restrict__ Dp,    const float* __restrict__ delta,
                     const float* __restrict__ dbias, const float* __restrict__ zg,
                     const float* __restrict__ Bt,    const float* __restrict__ Ct,
                     float* __restrict__ out) {
  __shared__ __align__(16) float sB[STEPS][NST];               // 16 KB
  __shared__ __align__(16) float sC[STEPS][NST];               // 16 KB
  __shared__ __align__(16) float sPart[WAVES_PER_BLOCK][16][34]; // ~17 KB, padded rows

  const int tid  = threadIdx.x;
  const int wave = tid >> 5;
  const int lane = tid & 31;
  const int b    = blockIdx.x >> 5;        // 32 workgroups per batch
  const int dgrp = blockIdx.x & 31;
  const int d    = dgrp * WAVES_PER_BLOCK + wave;

  // Per-wave row of A, 8 entries per lane, pre-scaled by log2(e) so the inner
  // loop uses native v_exp_f32 (base-2).
  const float LOG2E = 1.44269504088896340736f;
  float a2[8];
  {
    fvec4 a0 = *(const fvec4*)(A + (size_t)d * NST + lane * 8);
    fvec4 a1 = *(const fvec4*)(A + (size_t)d * NST + lane * 8 + 4);
#pragma unroll
    for (int k = 0; k < 4; ++k) { a2[k] = a0[k] * LOG2E; a2[k + 4] = a1[k] * LOG2E; }
  }
  float h[8];
#pragma unroll
  for (int k = 0; k < 8; ++k) h[k] = 0.f;

  const float Dpd = Dp[d];
  const int ph = d >> 4, pw = d & 15;                 // token -> patch coords
  const size_t bdl   = ((size_t)b * NF + d) * SL;     // delta/dbias/z base
  const float* BtB   = Bt + (size_t)b * SL * NST;
  const float* CtB   = Ct + (size_t)b * SL * NST;

  for (int blk = 0; blk < NBLK; ++blk) {
    const int l0 = blk * STEPS;
    __syncthreads();   // previous block's sB/sC fully consumed by all waves

    // Stage B/C columns for steps l0..l0+15: two contiguous 16 KB regions.
    {
      const float* gB = BtB + (size_t)l0 * NST;
      const float* gC = CtB + (size_t)l0 * NST;
      float* lB = &sB[0][0];
      float* lC = &sC[0][0];
#pragma unroll
      for (int t = 0; t < 4; ++t) {
        const int idx = (tid + t * 256) * 4;          // float index, 16B granules
        copy16_to_lds(gB + idx, lB + idx);
        copy16_to_lds(gC + idx, lC + idx);
      }
    }

    // Per-lane scalars for step l0 + (lane&15) — overlaps the async copies.
    const int l  = l0 + (lane & 15);
    const int c  = l >> 2, ii = (l >> 1) & 1, jj = l & 1;
    const int imaddr = ((b * PC + c) * IMG + (ph * 2 + ii)) * IMG + (pw * 2 + jj);
    const float u_l    = x[imaddr];
    const float dl_l   = softplus_f(delta[bdl + l] + dbias[bdl + l]);
    const float du_l   = dl_l * u_l;
    const float gate_l = silu_f(zg[bdl + l]);

    wait_async_copies();
    __syncthreads();

    float (*part)[34] = sPart[wave];
#pragma unroll
    for (int s = 0; s < STEPS; ++s) {
      const float dl = __shfl(dl_l, s, 32);
      const float du = __shfl(du_l, s, 32);
      const fvec4 blo = *(const fvec4*)&sB[s][lane * 8];
      const fvec4 bhi = *(const fvec4*)&sB[s][lane * 8 + 4];
      const fvec4 clo = *(const fvec4*)&sC[s][lane * 8];
      const fvec4 chi = *(const fvec4*)&sC[s][lane * 8 + 4];
      float accs = 0.f;
#pragma unroll
      for (int k = 0; k < 8; ++k) {
        const float bk = (k < 4) ? blo[k] : bhi[k - 4];
        const float ck = (k < 4) ? clo[k] : chi[k - 4];
        h[k] = fast_exp2(dl * a2[k]) * h[k] + du * bk;   // v_exp_f32 + v_fma
        accs += h[k] * ck;
      }
      part[s][lane] = accs;   // per-lane partial of y for step s
    }
    wave_fence();

    // Reduce 32 lane-partials for 16 steps at once.
    float y;
#ifdef HAVE_WMMA_F32X4
    {
      // D(m,n) = sum_k A(m,k)*B(k,n); with A==1 and B(k,n)=part[n][chunk(k)],
      // 8 accumulating WMMAs give D(m,n) = Y[n] for every m — so each lane
      // reads Y[lane&15] straight out of its own D register, no epilogue.
      v8f accv = {};
      v2f aones; aones[0] = 1.f; aones[1] = 1.f;
      const int prow = lane & 15;
      const int pcol = (lane >> 4) << 1;
#pragma unroll
      for (int ch = 0; ch < 8; ++ch) {
        v2f bp;
        bp[0] = part[prow][ch * 4 + pcol];
        bp[1] = part[prow][ch * 4 + pcol + 1];
        accv = __builtin_amdgcn_wmma_f32_16x16x4_f32(
            /*neg_a=*/false, aones, /*neg_b=*/false, bp,
            /*c_mod=*/(short)0, accv, /*reuse_a=*/false, /*reuse_b=*/false);
      }
      y = accv[0];
    }
#else
    {
      float sum = 0.f;
#pragma unroll
      for (int j = 0; j < 32; ++j) sum += part[lane & 15][j];
      y = sum;
    }
#endif

    const float outv = (y + u_l * Dpd) * gate_l;
    if (lane < 16) out[imaddr] = outv;   // fold back into image layout
  }
}

// ---------------------------------------------------------------------------
extern "C" void kernel_launch(void* const* d_in, const int* in_sizes, int n_in,
                              void* d_out, int out_size, void* d_ws, size_t ws_size,
                              hipStream_t stream) {
  const float* x  = (const float*)d_in[0];   // (8,64,32,32)
  const float* A  = (const float*)d_in[1];   // (256,256)
  const float* Bp = (const float*)d_in[2];   // (8,256,256)
  const float* Cp = (const float*)d_in[3];   // (8,256,256)
  const float* Dp = (const float*)d_in[4];   // (256,)
  const float* dl = (const float*)d_in[5];   // (8,256,256)
  const float* db = (const float*)d_in[6];   // (8,256,256)
  const float* zg = (const float*)d_in[7];   // (8,256,256)
  float* out = (float*)d_out;

  // Workspace: Bt and Ct in [b][l][n] layout, 2 MB each.
  float* Bt = (float*)d_ws;
  float* Ct = Bt + (size_t)BATCH * SL * NST;

  dim3 tgrid(SL / 32, NF / 32, BATCH * 2);
  transpose_bc_kernel<<<tgrid, dim3(32, 8), 0, stream>>>(Bp, Cp, Bt, Ct);

  ssm_scan_kernel<<<BATCH * (NF / WAVES_PER_BLOCK), 256, 0, stream>>>(
      x, A, Dp, dl, db, zg, Bt, Ct, out);
  (void)in_sizes; (void)n_in; (void)out_size; (void)ws_size;
}